// GraphConv_37709812859405
// MI455X (gfx1250) — compile-verified
//
#include <hip/hip_runtime.h>

typedef __attribute__((ext_vector_type(2))) float v2f;
typedef __attribute__((ext_vector_type(8))) float v8f;

#define DIMF 128
#define LDS_PITCH 132   // 128 + 4 pad -> conflict-free ds_load_b64 fragment reads

// ---------------------------------------------------------------------------
// Kernel 1: h2 = h @ W.T + b   (full-f32 WMMA, 16x16x4)
// Also seeds out = h2 (residual term / atomic accumulator init).
// Block = 256 threads = 8 waves; block covers 128 rows; wave covers 16 rows.
// ---------------------------------------------------------------------------
__global__ __launch_bounds__(256) void linear_wmma_kernel(
    const float* __restrict__ h, const float* __restrict__ W,
    const float* __restrict__ bias, float* __restrict__ h2,
    float* __restrict__ out, int n_nodes)
{
    __shared__ __align__(16) float sW[DIMF * LDS_PITCH];
    __shared__ __align__(16) float sH[DIMF * LDS_PITCH];

    const int tid      = threadIdx.x;
    const int rowBlock = blockIdx.x * 128;

    // Cooperative staging: W (64 KB) and one 128-row slab of h, padded pitch.
    // 4096 float4 slots each, 16 per thread, fully coalesced global_load_b128.
#pragma unroll
    for (int i = 0; i < 16; ++i) {
        const int slot = tid + i * 256;
        const int r  = slot >> 5;     // row 0..127
        const int c4 = slot & 31;     // float4 column 0..31
        const float4 w4 = ((const float4*)W)[slot];
        *(float4*)&sW[r * LDS_PITCH + c4 * 4] = w4;
        const int gr = rowBlock + r;
        float4 h4 = make_float4(0.f, 0.f, 0.f, 0.f);
        if (gr < n_nodes)
            h4 = ((const float4*)h)[(size_t)gr * (DIMF / 4) + c4];
        *(float4*)&sH[r * LDS_PITCH + c4 * 4] = h4;
    }
    __syncthreads();

    const int wave = tid >> 5;
    const int lane = tid & 31;
    const int tileRow = rowBlock + wave * 16;   // n_nodes % 16 == 0 -> no partial tiles
    if (tileRow >= n_nodes) return;             // uniform per wave: EXEC stays all-1s

    const int n  = lane & 15;   // N index within tile / M index for A frag
    const int hh = lane >> 4;   // half-wave selector (K pair / M+8 for C)

    v8f acc[8];
    const v8f vzero = {0.f, 0.f, 0.f, 0.f, 0.f, 0.f, 0.f, 0.f};
#pragma unroll
    for (int nt = 0; nt < 8; ++nt) acc[nt] = vzero;

    // A frag (16x4 f32): lane 16*hh+m holds {A[m][kk+2hh], A[m][kk+2hh+1]}
    // B frag (4x16 f32): lane 16*hh+n holds {B[2hh][n], B[2hh+1][n]} = W[n_g][kk+2hh..+1]
    const float* aBase = &sH[(wave * 16 + n) * LDS_PITCH + 2 * hh];
    const float* bBase = &sW[n * LDS_PITCH + 2 * hh];

#pragma unroll 8
    for (int kk = 0; kk < DIMF; kk += 4) {
        const v2f a = *(const v2f*)(aBase + kk);
#pragma unroll
        for (int nt = 0; nt < 8; ++nt) {
            const v2f bf = *(const v2f*)(bBase + nt * 16 * LDS_PITCH + kk);
            acc[nt] = __builtin_amdgcn_wmma_f32_16x16x4_f32(
                false, a, false, bf, (short)0, acc[nt], false, false);
        }
    }

    // C/D layout: VGPR r -> row (r + 8*hh), lanes 0..15 -> col n
    const int gRow0 = tileRow + hh * 8;
#pragma unroll
    for (int nt = 0; nt < 8; ++nt) {
        const int col = nt * 16 + n;
        const float bv = bias[col];
#pragma unroll
        for (int r = 0; r < 8; ++r) {
            const float vo = acc[nt][r] + bv;
            const size_t off = (size_t)(gRow0 + r) * DIMF + col;
            h2[off]  = vo;   // gather source for SpMM
            out[off] = vo;   // residual term = atomic accumulator init
        }
    }
}

// ---------------------------------------------------------------------------
// Kernel 2: out[row[e]] += val[e] * h2[col[e]]   (one wave32 per edge)
// float4 gather (512 B coalesced per wave) + hw f32 atomics into L2.
// ---------------------------------------------------------------------------
__global__ __launch_bounds__(256) void spmm_edge_kernel(
    const int* __restrict__ erow, const int* __restrict__ ecol,
    const float* __restrict__ eval, const float* __restrict__ x,
    float* __restrict__ out, int n_edges)
{
    const int e = (int)((blockIdx.x * 256u + threadIdx.x) >> 5);
    if (e >= n_edges) return;
    const int lane = threadIdx.x & 31;

    const int   r = erow[e];
    const int   c = ecol[e];
    const float v = eval[e];

    const float4 g = ((const float4*)(x + (size_t)c * DIMF))[lane];
    float* op = out + (size_t)r * DIMF + lane * 4;

    unsafeAtomicAdd(op + 0, v * g.x);   // global_atomic_add_f32
    unsafeAtomicAdd(op + 1, v * g.y);
    unsafeAtomicAdd(op + 2, v * g.z);
    unsafeAtomicAdd(op + 3, v * g.w);
}

// ---------------------------------------------------------------------------
// Kernel 3: in-place ReLU (float4)
// ---------------------------------------------------------------------------
__global__ __launch_bounds__(256) void relu_kernel(float* __restrict__ out, int n4)
{
    const int i = blockIdx.x * blockDim.x + threadIdx.x;
    if (i >= n4) return;
    float4* p = (float4*)out;
    float4 v = p[i];
    v.x = fmaxf(v.x, 0.f);
    v.y = fmaxf(v.y, 0.f);
    v.z = fmaxf(v.z, 0.f);
    v.w = fmaxf(v.w, 0.f);
    p[i] = v;
}

extern "C" void kernel_launch(void* const* d_in, const int* in_sizes, int n_in,
                              void* d_out, int out_size, void* d_ws, size_t ws_size,
                              hipStream_t stream)
{
    const float* h    = (const float*)d_in[0];
    const float* W    = (const float*)d_in[1];
    const float* b    = (const float*)d_in[2];
    const float* eval = (const float*)d_in[3];
    const int*   erow = (const int*)d_in[4];
    const int*   ecol = (const int*)d_in[5];
    float* out = (float*)d_out;
    float* h2  = (float*)d_ws;          // 50000*128*4 = 25.6 MB scratch

    const int n_nodes = in_sizes[0] / DIMF;
    const int n_edges = in_sizes[3];

    const int blocks1 = (n_nodes + 127) / 128;
    linear_wmma_kernel<<<blocks1, 256, 0, stream>>>(h, W, b, h2, out, n_nodes);

    const long long spmm_threads = (long long)n_edges * 32;
    const int blocks2 = (int)((spmm_threads + 255) / 256);
    spmm_edge_kernel<<<blocks2, 256, 0, stream>>>(erow, ecol, eval, h2, out, n_edges);

    const int n4 = n_nodes * DIMF / 4;
    relu_kernel<<<(n4 + 255) / 256, 256, 0, stream>>>(out, n4);
}